// Network_54735063220774
// MI455X (gfx1250) — compile-verified
//
#include <hip/hip_runtime.h>
#include <hip/hip_bf16.h>

#define AS1 __attribute__((address_space(1)))
#define AS3 __attribute__((address_space(3)))

typedef __attribute__((ext_vector_type(2))) float v2f;
typedef __attribute__((ext_vector_type(8))) float v8f;
typedef __attribute__((ext_vector_type(4))) int   v4i;

// Problem constants (fixed by the reference)
#define BATCH   16384
#define TSTEPS  2048
#define BLOCK   128          // threads per block = rows per block (1 row per lane)
#define NBLK    (BATCH / BLOCK)   // 128 blocks
#define TK      32           // time-tile width (floats per row per tile)
#define NT      (TSTEPS / TK)     // 64 tiles
#define LROW    36           // padded LDS row stride (16B-aligned rows, conflict-broken)
#define NBUF    3            // triple buffer: 2-deep async prefetch

// ---- CDNA5 async global->LDS copy (ASYNCcnt path, CDNA5 ISA) ----
__device__ __forceinline__ void async_b128(const float* gp, float* lp) {
#if __has_builtin(__builtin_amdgcn_global_load_async_to_lds_b128)
    __builtin_amdgcn_global_load_async_to_lds_b128(
        (AS1 v4i*)gp, (AS3 v4i*)lp, 0, 0);
#else
    unsigned loff = (unsigned)(unsigned long long)(AS3 char*)lp;
    asm volatile("global_load_async_to_lds_b128 %0, %1, off"
                 :: "v"(loff), "v"(gp) : "memory");
#endif
}

template <int N>
__device__ __forceinline__ void wait_async() {
#if __has_builtin(__builtin_amdgcn_s_wait_asynccnt)
    __builtin_amdgcn_s_wait_asynccnt(N);
#else
    asm volatile("s_wait_asynccnt %0" :: "n"(N) : "memory");
#endif
}

// Stage one 128-row x TK-col tile of x into LDS, coalesced (8 lanes cover one
// 128B row-chunk). 8 chunks of b128 per thread.
__device__ __forceinline__ void stage_tile(const float* gbase, float* lbuf, int tid) {
#pragma unroll
    for (int p = 0; p < 8; ++p) {
        int idx = p * BLOCK + tid;
        int r = idx >> 3;          // row 0..127
        int c = idx & 7;           // 16B chunk 0..7 within row
        async_b128(gbase + (size_t)r * TSTEPS + c * 4,
                   lbuf + r * LROW + c * 4);
    }
}

// Fast tanh on the TRANS unit.
__device__ __forceinline__ float ftanh(float x) {
#if __has_builtin(__builtin_amdgcn_tanhf)
    return __builtin_amdgcn_tanhf(x);
#else
    float cx = fminf(fmaxf(x, -12.0f), 12.0f);
    float e  = __builtin_amdgcn_exp2f(cx * 2.8853900817779268f); // 2*log2(e)
    return (e - 1.0f) * __builtin_amdgcn_rcpf(e + 1.0f);
#endif
}

__global__ __launch_bounds__(BLOCK) void rnn_scan_kernel(
    const float* __restrict__ x,   // [B,T]
    const float* __restrict__ y,   // [B]
    const float* __restrict__ wih, // [3]
    const float* __restrict__ whh, // [3,3] row-major
    const float* __restrict__ who, // [3]
    float* __restrict__ pred_out,  // [B] (d_out+1)
    float* __restrict__ ws)        // [NBLK] partial sq-err sums
{
    __shared__ float tile[NBUF][BLOCK * LROW];
    __shared__ float red[BLOCK / 32];

    const int tid = threadIdx.x;
    const int rowBase = blockIdx.x * BLOCK;
    const int row = rowBase + tid;
    const float* xrow_base = x + (size_t)rowBase * TSTEPS;

    // Uniform weights -> scalar loads
    const float wi0 = wih[0], wi1 = wih[1], wi2 = wih[2];
    const float a00 = whh[0], a01 = whh[1], a02 = whh[2];
    const float a10 = whh[3], a11 = whh[4], a12 = whh[5];
    const float a20 = whh[6], a21 = whh[7], a22 = whh[8];
    const float o0 = who[0], o1 = who[1], o2 = who[2];

    float h0 = 0.0f, h1 = 0.0f, h2 = 0.0f;

    // Prime the async pipeline: 2 tiles in flight.
    stage_tile(xrow_base + 0 * TK, tile[0], tid);
    stage_tile(xrow_base + 1 * TK, tile[1], tid);

    for (int t = 0; t < NT; ++t) {
        if (t + 2 < NT)
            stage_tile(xrow_base + (size_t)(t + 2) * TK, tile[(t + 2) % NBUF], tid);

        const int rem = NT - 1 - t;              // tiles still in flight beyond current
        if (rem >= 2)      wait_async<16>();     // keep 2 tiles (16 ops) outstanding
        else if (rem == 1) wait_async<8>();
        else               wait_async<0>();
        __syncthreads();                         // tile[t%NBUF] visible to all waves

        const float* srow = &tile[t % NBUF][tid * LROW];
#pragma unroll
        for (int tt = 0; tt < TK; ++tt) {
            float xv = srow[tt];                 // ds_load_b32
            float u0 = fmaf(xv, wi0, fmaf(h0, a00, fmaf(h1, a10, h2 * a20)));
            float u1 = fmaf(xv, wi1, fmaf(h0, a01, fmaf(h1, a11, h2 * a21)));
            float u2 = fmaf(xv, wi2, fmaf(h0, a02, fmaf(h1, a12, h2 * a22)));
            h0 = ftanh(u0);
            h1 = ftanh(u1);
            h2 = ftanh(u2);
        }
        __syncthreads();                         // all waves done reading before refill
    }

    // Final projection + per-row squared error.
    float pred = fmaf(h0, o0, fmaf(h1, o1, h2 * o2));
    pred_out[row] = pred;
    float diff = y[row] - pred;
    float sq = diff * diff;

    // ---- Wave-level sum of 32 squared errors via one WMMA ----
    // A: each lane contributes sq in one A-slot (VGPR0), zero in the other.
    // B = all-ones  =>  every column of D = sum over ALL A slots = wave total,
    // independent of the exact A slot mapping. Then each lane sums its 8 D
    // regs (half a column) and shfl_xor(16) completes the column.
    v2f a; a.x = sq;  a.y = 0.0f;
    v2f b; b.x = 1.0f; b.y = 1.0f;
    v8f c = {};
    c = __builtin_amdgcn_wmma_f32_16x16x4_f32(
            /*neg_a=*/false, a, /*neg_b=*/false, b,
            /*c_mod=*/(short)0, c, /*reuse_a=*/false, /*reuse_b=*/false);
    float p = c[0] + c[1] + c[2] + c[3] + c[4] + c[5] + c[6] + c[7];
    p += __shfl_xor(p, 16);                      // full wave sum in every lane

    if ((tid & 31) == 0) red[tid >> 5] = p;
    __syncthreads();
    if (tid == 0) {
        float s = 0.0f;
#pragma unroll
        for (int w = 0; w < BLOCK / 32; ++w) s += red[w];
        ws[blockIdx.x] = s;
    }
}

__global__ __launch_bounds__(NBLK) void loss_reduce_kernel(
    const float* __restrict__ ws, float* __restrict__ out)
{
    int tid = threadIdx.x;                       // NBLK = 128 threads
    float v = ws[tid];
#pragma unroll
    for (int o = 16; o > 0; o >>= 1) v += __shfl_down(v, o);
    __shared__ float s[NBLK / 32];
    if ((tid & 31) == 0) s[tid >> 5] = v;
    __syncthreads();
    if (tid == 0) {
        float acc = 0.0f;
#pragma unroll
        for (int w = 0; w < NBLK / 32; ++w) acc += s[w];
        out[0] = acc * (1.0f / (float)BATCH);    // mean squared error
    }
}

extern "C" void kernel_launch(void* const* d_in, const int* in_sizes, int n_in,
                              void* d_out, int out_size, void* d_ws, size_t ws_size,
                              hipStream_t stream) {
    (void)in_sizes; (void)n_in; (void)out_size; (void)ws_size;
    const float* x   = (const float*)d_in[0];  // [16384, 2048]
    const float* y   = (const float*)d_in[1];  // [16384]
    const float* wih = (const float*)d_in[2];  // [1,3]
    const float* whh = (const float*)d_in[3];  // [3,3]
    const float* who = (const float*)d_in[4];  // [3,1]
    float* out = (float*)d_out;                // [0]=loss, [1..B]=pred
    float* ws  = (float*)d_ws;                 // NBLK partial sums

    rnn_scan_kernel<<<dim3(NBLK), dim3(BLOCK), 0, stream>>>(
        x, y, wih, whh, who, out + 1, ws);
    loss_reduce_kernel<<<dim3(1), dim3(NBLK), 0, stream>>>(ws, out);
}